// Attention_30253749633178
// MI455X (gfx1250) — compile-verified
//
#include <hip/hip_runtime.h>
#include <math.h>

// Problem constants (reference: DIM=512, HEADS=8, HEAD_DIM=64, B=4, N=256)
#define DIMC     512
#define BATCH    4
#define SEQ      256
#define HEADS    8
#define HEAD_DIM 64
#define SCALE    0.125f

typedef __attribute__((ext_vector_type(2))) float v2f;
typedef __attribute__((ext_vector_type(8))) float v8f;

// D = A(16x4 f32) * B(4x16 f32) + C(16x16 f32), wave32 WMMA.
__device__ __forceinline__ v8f wmma4(v2f a, v2f b, v8f c) {
    return __builtin_amdgcn_wmma_f32_16x16x4_f32(
        /*neg_a=*/false, a, /*neg_b=*/false, b,
        /*c_mod=*/(short)0, c, /*reuse_a=*/false, /*reuse_b=*/false);
}

// ---------------------------------------------------------------------------
// Kernel 1: fused Q/K/V projection.  Y = X @ W^T  (X: [1024,512], W: [512,512])
// One wave computes a 16x64 output tile (A fragment shared by 4 N-subtiles).
// grid = 384 blocks * 128 threads = 1536 waves = 3 proj * 512 tiles.
// ---------------------------------------------------------------------------
__global__ __launch_bounds__(128)
void proj_kernel(const float* __restrict__ q, const float* __restrict__ k,
                 const float* __restrict__ v,
                 const float* __restrict__ Wq, const float* __restrict__ Wk,
                 const float* __restrict__ Wv,
                 float* __restrict__ qh, float* __restrict__ kh,
                 float* __restrict__ vh) {
    const int lane = threadIdx.x & 31;
    const int wave = threadIdx.x >> 5;
    const int gw   = blockIdx.x * 4 + wave;     // 0..1535
    const int p    = gw >> 9;                   // 0=q,1=k,2=v
    const int tid  = gw & 511;                  // tile id within projection
    const int tm   = tid & 63;                  // row-16 tile   (1024/16)
    const int tn   = tid >> 6;                  // col-64 tile   (512/64)

    const float* X = (p == 0) ? q  : (p == 1) ? k  : v;
    const float* W = (p == 0) ? Wq : (p == 1) ? Wk : Wv;
    float*       Y = (p == 0) ? qh : (p == 1) ? kh : vh;

    const int lrow = lane & 15;
    const int koff = (lane >> 4) << 1;          // 0 or 2

    const float* arow = X + (size_t)(tm * 16 + lrow) * DIMC + koff;
    const float* brow0 = W + (size_t)(tn * 64 + 0 * 16 + lrow) * DIMC + koff;
    const float* brow1 = W + (size_t)(tn * 64 + 1 * 16 + lrow) * DIMC + koff;
    const float* brow2 = W + (size_t)(tn * 64 + 2 * 16 + lrow) * DIMC + koff;
    const float* brow3 = W + (size_t)(tn * 64 + 3 * 16 + lrow) * DIMC + koff;

    v8f acc0 = {}, acc1 = {}, acc2 = {}, acc3 = {};
    #pragma unroll 8
    for (int k0 = 0; k0 < DIMC; k0 += 4) {
        v2f a  = *(const v2f*)(arow  + k0);
        v2f b0 = *(const v2f*)(brow0 + k0);
        v2f b1 = *(const v2f*)(brow1 + k0);
        v2f b2 = *(const v2f*)(brow2 + k0);
        v2f b3 = *(const v2f*)(brow3 + k0);
        acc0 = wmma4(a, b0, acc0);
        acc1 = wmma4(a, b1, acc1);
        acc2 = wmma4(a, b2, acc2);
        acc3 = wmma4(a, b3, acc3);
    }

    // C/D layout: VGPR vv -> row (vv + (lane>=16 ? 8 : 0)), col = lane%16.
    const int rbase = tm * 16 + ((lane >> 4) << 3);
    #pragma unroll
    for (int vv = 0; vv < 8; ++vv) {
        float* yr = Y + (size_t)(rbase + vv) * DIMC + tn * 64 + lrow;
        yr[0]  = acc0[vv];
        yr[16] = acc1[vv];
        yr[32] = acc2[vv];
        yr[48] = acc3[vv];
    }
}

// ---------------------------------------------------------------------------
// Kernel 2: per-(b,h) attention.  One wave owns a 16-row query tile:
//   S = Qh @ Kh^T (16x256 in regs), softmax rows via half-wave shfl_xor,
//   write attn, then X = attn @ Vh (reload attn as A-fragments; hits L2).
// grid = B*H*4 = 128 blocks * 128 threads (4 waves) -> 16 row-tiles per (b,h).
// ---------------------------------------------------------------------------
__global__ __launch_bounds__(128)
void attn_kernel(const float* __restrict__ qh, const float* __restrict__ kh,
                 const float* __restrict__ vh,
                 float* __restrict__ x_out, float* __restrict__ attn_out) {
    const int lane = threadIdx.x & 31;
    const int wave = threadIdx.x >> 5;
    const int bh   = blockIdx.x >> 2;            // 0..31
    const int b    = bh >> 3;
    const int h    = bh & 7;
    const int t1   = ((blockIdx.x & 3) << 2) + wave; // q row-tile 0..15
    const int n1b  = t1 * 16;

    const int lrow = lane & 15;
    const int koff = (lane >> 4) << 1;

    const size_t hb = (size_t)b * SEQ * DIMC + (size_t)h * HEAD_DIM;
    const float* Q = qh + hb;
    const float* K = kh + hb;
    const float* V = vh + hb;
    float* attn = attn_out + (size_t)bh * SEQ * SEQ;

    // Preload Q fragments for the whole K-dim (HS=64 -> 16 steps of 4).
    v2f aq[16];
    #pragma unroll
    for (int kk = 0; kk < 16; ++kk)
        aq[kk] = *(const v2f*)(Q + (size_t)(n1b + lrow) * DIMC + kk * 4 + koff);

    // S = Q K^T : 16 tiles of 16x16 along N2.
    v8f S[16];
    #pragma unroll
    for (int t2 = 0; t2 < 16; ++t2) {
        const float* brow = K + (size_t)(t2 * 16 + lrow) * DIMC + koff;
        v8f acc = {};
        #pragma unroll
        for (int kk = 0; kk < 16; ++kk)
            acc = wmma4(aq[kk], *(const v2f*)(brow + kk * 4), acc);
        S[t2] = acc;
    }

    // Row softmax.  Row r of a C-tile lives in VGPR (r%8) across one 16-lane
    // half; xor masks 1,2,4,8 reduce within that half.
    #pragma unroll
    for (int vv = 0; vv < 8; ++vv) {
        float mx = -3.402823466e38f;
        #pragma unroll
        for (int t2 = 0; t2 < 16; ++t2) mx = fmaxf(mx, S[t2][vv]);
        #pragma unroll
        for (int m = 1; m < 16; m <<= 1) mx = fmaxf(mx, __shfl_xor(mx, m, 32));
        mx *= SCALE;

        float sum = 0.f;
        #pragma unroll
        for (int t2 = 0; t2 < 16; ++t2) {
            float e = __expf(S[t2][vv] * SCALE - mx);
            S[t2][vv] = e;
            sum += e;
        }
        #pragma unroll
        for (int m = 1; m < 16; m <<= 1) sum += __shfl_xor(sum, m, 32);

        const float inv = __frcp_rn(sum);
        #pragma unroll
        for (int t2 = 0; t2 < 16; ++t2) S[t2][vv] *= inv;
    }

    // Write attn [bh, n1, n2].
    const int rbase = n1b + ((lane >> 4) << 3);
    #pragma unroll
    for (int t2 = 0; t2 < 16; ++t2)
        #pragma unroll
        for (int vv = 0; vv < 8; ++vv)
            attn[(size_t)(rbase + vv) * SEQ + t2 * 16 + lrow] = S[t2][vv];

    // X = attn @ Vh : reload attn rows as A fragments (same wave wrote them;
    // HW keeps store->load same-address in order; data is hot in L2).
    const float* arow = attn + (size_t)(n1b + lrow) * SEQ + koff;
    #pragma unroll 1
    for (int dt = 0; dt < 4; ++dt) {               // 4 tiles of 16 head dims
        const float* vcol = V + (size_t)dt * 16 + lrow;
        v8f acc = {};
        #pragma unroll 4
        for (int k0 = 0; k0 < SEQ; k0 += 4) {
            v2f a = *(const v2f*)(arow + k0);
            const int kr = k0 + koff;
            v2f bb = { vcol[(size_t)kr * DIMC], vcol[(size_t)(kr + 1) * DIMC] };
            acc = wmma4(a, bb, acc);
        }
        #pragma unroll
        for (int vv = 0; vv < 8; ++vv)
            x_out[((size_t)b * SEQ + rbase + vv) * DIMC +
                  (size_t)h * HEAD_DIM + dt * 16 + lrow] = acc[vv];
    }
}

// ---------------------------------------------------------------------------
extern "C" void kernel_launch(void* const* d_in, const int* in_sizes, int n_in,
                              void* d_out, int out_size, void* d_ws, size_t ws_size,
                              hipStream_t stream) {
    const float* q  = (const float*)d_in[0];
    const float* k  = (const float*)d_in[1];
    const float* v  = (const float*)d_in[2];
    // d_in[3] relation_feature: results discarded by reference -> never read.
    const float* Wq = (const float*)d_in[4];
    const float* Wk = (const float*)d_in[5];
    const float* Wv = (const float*)d_in[6];
    // d_in[7] W_r_conv, d_in[8] W_r_qk: unused.

    const size_t proj_elems = (size_t)BATCH * SEQ * DIMC;   // 524288
    float* qh = (float*)d_ws;
    float* kh = qh + proj_elems;
    float* vh = kh + proj_elems;

    float* x    = (float*)d_out;                 // [B,N,C]
    float* attn = x + proj_elems;                // [B,H,N,N]

    proj_kernel<<<384, 128, 0, stream>>>(q, k, v, Wq, Wk, Wv, qh, kh, vh);
    attn_kernel<<<128, 128, 0, stream>>>(qh, kh, vh, x, attn);
}